// SMPLlayer_46273977647579
// MI455X (gfx1250) — compile-verified
//
#include <hip/hip_runtime.h>
#include <hip/hip_bf16.h>

typedef __attribute__((ext_vector_type(2))) float v2f;
typedef __attribute__((ext_vector_type(8))) float v8f;

#define NV   6890
#define NV3  (NV * 3)
#define NJ   24
#define NB   10
#define NP   207

__device__ const int kPar[24] = {-1, 0, 0, 0, 1, 2, 3, 4, 5, 6, 7, 8,
                                  9, 9, 9, 12, 13, 14, 16, 17, 18, 19, 20, 21};

// ---------------------------------------------------------------------------
// Kernel 1: v_shaped[i] = v_template[i] + sum_l shapes0[l] * shapedirs[i][l]
// i runs over V*3; shapedirs is (V,3,NB) row-major -> [i*NB + l]
// ---------------------------------------------------------------------------
__global__ void k_vshaped(const float* __restrict__ vt,
                          const float* __restrict__ sd,
                          const float* __restrict__ shapes,
                          float* __restrict__ vs) {
    int i = blockIdx.x * blockDim.x + threadIdx.x;
    if (i >= NV3) return;
    float acc = vt[i];
#pragma unroll
    for (int l = 0; l < NB; ++l) acc += shapes[l] * sd[i * NB + l];
    vs[i] = acc;
}

// ---------------------------------------------------------------------------
// Kernel 2: J[j][t] = sum_v J_regressor[j][v] * v_shaped[v][t]
// One block per joint, 256-thread reduction.
// ---------------------------------------------------------------------------
__global__ void k_joints(const float* __restrict__ JR,
                         const float* __restrict__ vs,
                         float* __restrict__ Jout) {
    __shared__ float red[3 * 256];
    int j = blockIdx.x;
    float s0 = 0.f, s1 = 0.f, s2 = 0.f;
    for (int v = threadIdx.x; v < NV; v += 256) {
        float w = JR[j * NV + v];
        s0 += w * vs[v * 3 + 0];
        s1 += w * vs[v * 3 + 1];
        s2 += w * vs[v * 3 + 2];
    }
    red[threadIdx.x] = s0;
    red[256 + threadIdx.x] = s1;
    red[512 + threadIdx.x] = s2;
    __syncthreads();
    for (int off = 128; off > 0; off >>= 1) {
        if (threadIdx.x < off) {
            red[threadIdx.x]       += red[threadIdx.x + off];
            red[256 + threadIdx.x] += red[256 + threadIdx.x + off];
            red[512 + threadIdx.x] += red[512 + threadIdx.x + off];
        }
        __syncthreads();
    }
    if (threadIdx.x == 0) {
        Jout[j * 3 + 0] = red[0];
        Jout[j * 3 + 1] = red[256];
        Jout[j * 3 + 2] = red[512];
    }
}

// ---------------------------------------------------------------------------
// Rodrigues: axis-angle -> 3x3 (matches reference: I + s*K + (1-c)*K^2,
// with K^2 = aa^T - I for the normalized axis)
// ---------------------------------------------------------------------------
__device__ inline void rodrigues3(float r0, float r1, float r2, float* R) {
    float ang = sqrtf(r0 * r0 + r1 * r1 + r2 * r2 + 1e-8f);
    float inv = 1.f / ang;
    float ax = r0 * inv, ay = r1 * inv, az = r2 * inv;
    float c = cosf(ang), s = sinf(ang), t = 1.f - c;
    R[0] = 1.f + t * (ax * ax - 1.f);
    R[1] = -s * az + t * ax * ay;
    R[2] =  s * ay + t * ax * az;
    R[3] =  s * az + t * ax * ay;
    R[4] = 1.f + t * (ay * ay - 1.f);
    R[5] = -s * ax + t * ay * az;
    R[6] = -s * ay + t * ax * az;
    R[7] =  s * ax + t * ay * az;
    R[8] = 1.f + t * (az * az - 1.f);
}

// ---------------------------------------------------------------------------
// Kernel 3 (one wave): rotations, pose_feature, kinematic chain, A' = rotG @ A_rel
// A' stored as WMMA-B-ready rows of 16: Aws[j*16 + r*4 + c] (n=12..15 zero)
// ---------------------------------------------------------------------------
__global__ void k_chain(const float* __restrict__ poses,
                        const float* __restrict__ Rh,
                        const float* __restrict__ Jws,
                        float* __restrict__ pf,
                        float* __restrict__ Aws) {
    __shared__ float rot[24][9];
    __shared__ float rotG[9];
    __shared__ float rel[24][3];
    __shared__ float Rw[24][9];
    __shared__ float tw[24][3];
    int t = threadIdx.x;
    if (t < 24) rodrigues3(poses[3 * t], poses[3 * t + 1], poses[3 * t + 2], rot[t]);
    if (t == 31) rodrigues3(Rh[0], Rh[1], Rh[2], rotG);
    __syncthreads();
    if (t >= 1 && t < 24) {
#pragma unroll
        for (int k = 0; k < 9; ++k)
            pf[(t - 1) * 9 + k] = rot[t][k] - ((k % 4 == 0) ? 1.f : 0.f);
    }
    if (t < 24) {
        int p = kPar[t];
#pragma unroll
        for (int c = 0; c < 3; ++c)
            rel[t][c] = Jws[t * 3 + c] - (p >= 0 ? Jws[p * 3 + c] : 0.f);
    }
    __syncthreads();
    if (t == 0) {  // serial tree composition (24 tiny matmuls)
#pragma unroll
        for (int k = 0; k < 9; ++k) Rw[0][k] = rot[0][k];
#pragma unroll
        for (int c = 0; c < 3; ++c) tw[0][c] = rel[0][c];
        for (int j = 1; j < 24; ++j) {
            int p = kPar[j];
            for (int r = 0; r < 3; ++r) {
                for (int c = 0; c < 3; ++c) {
                    float acc = 0.f;
                    for (int k = 0; k < 3; ++k) acc += Rw[p][r * 3 + k] * rot[j][k * 3 + c];
                    Rw[j][r * 3 + c] = acc;
                }
                float acc = tw[p][r];
                for (int k = 0; k < 3; ++k) acc += Rw[p][r * 3 + k] * rel[j][k];
                tw[j][r] = acc;
            }
        }
    }
    __syncthreads();
    if (t < 24) {
        float tr[3];
#pragma unroll
        for (int r = 0; r < 3; ++r) {
            float acc = tw[t][r];
            for (int k = 0; k < 3; ++k) acc -= Rw[t][r * 3 + k] * Jws[t * 3 + k];
            tr[r] = acc;
        }
#pragma unroll
        for (int r = 0; r < 3; ++r) {
            for (int c = 0; c < 3; ++c) {
                float acc = 0.f;
                for (int k = 0; k < 3; ++k) acc += rotG[r * 3 + k] * Rw[t][k * 3 + c];
                Aws[t * 16 + r * 4 + c] = acc;
            }
            float acc = 0.f;
            for (int k = 0; k < 3; ++k) acc += rotG[r * 3 + k] * tr[k];
            Aws[t * 16 + r * 4 + 3] = acc;
        }
#pragma unroll
        for (int n = 12; n < 16; ++n) Aws[t * 16 + n] = 0.f;
    }
}

// ---------------------------------------------------------------------------
// Kernel 4: per 16-vertex tile (one wave each):
//   T(16x16) = weights_tile(16x24) @ A'(24x16) via 6x V_WMMA_F32_16X16X4_F32
//   v_posed = v_shaped + pose_feature @ posedirs (per-tile GEMV)
//   out = T[:3,:3] @ v_posed + T[:3,3] + Th
// Tail handling is branchless: vertex index clamped to NV-1 for all loads
// (duplicate valid rows), only the final global store is guarded.
// ---------------------------------------------------------------------------
__global__ __launch_bounds__(128) void k_lbs(const float* __restrict__ W,
                                             const float* __restrict__ Aws,
                                             const float* __restrict__ pf,
                                             const float* __restrict__ PD,
                                             const float* __restrict__ vs,
                                             const float* __restrict__ Th,
                                             float* __restrict__ out,
                                             int nTiles) {
    __shared__ float smem[4][256 + 48];
    int wave = threadIdx.x >> 5;
    int lane = threadIdx.x & 31;
    int tile = blockIdx.x * 4 + wave;
    if (tile >= nTiles) return;  // wave-uniform: EXEC stays all-ones for WMMA
    int vbase = tile * 16;
    int row = lane & 15;
    int khalf = (lane >= 16) ? 2 : 0;
    int vr = vbase + row;
    int vrc = vr < NV ? vr : (NV - 1);  // branchless tail clamp

    v8f acc = {0.f, 0.f, 0.f, 0.f, 0.f, 0.f, 0.f, 0.f};
#pragma unroll
    for (int jb = 0; jb < 24; jb += 4) {
        v2f a, b;
        // A (16x4 f32): v0 = A[m][khalf], v1 = A[m][khalf+1]
        a.x = W[vrc * 24 + jb + khalf];
        a.y = W[vrc * 24 + jb + khalf + 1];
        // B (4x16 f32): v0 = B[khalf][n], v1 = B[khalf+1][n]
        b.x = Aws[(jb + khalf) * 16 + row];
        b.y = Aws[(jb + khalf + 1) * 16 + row];
        acc = __builtin_amdgcn_wmma_f32_16x16x4_f32(
            false, a, false, b, (short)0, acc, false, false);
    }

    float* sTv = smem[wave];        // 16x16 blended transforms
    float* svp = smem[wave] + 256;  // 16x3 posed vertices
    int mbase = (lane >= 16) ? 8 : 0;
#pragma unroll
    for (int i = 0; i < 8; ++i)
        sTv[(i + mbase) * 16 + row] = acc[i];

    // pose blendshape GEMV for this tile's vertices: 48 (vertex,comp) tasks
    for (int task = lane; task < 48; task += 32) {
        int m = task / 3, tt = task % 3;
        int v = vbase + m;
        int vc = v < NV ? v : (NV - 1);  // branchless tail clamp
        float d = vs[vc * 3 + tt];
        for (int p = 0; p < NP; ++p)
            d += pf[p] * PD[p * NV3 + vc * 3 + tt];
        svp[task] = d;
    }
    // same-wave LDS ops are in-order; wait for outstanding DS writes anyway
    asm volatile("s_wait_dscnt 0" ::: "memory");

    for (int task = lane; task < 48; task += 32) {
        int m = task / 3, r = task % 3;
        int v = vbase + m;
        if (v < NV) {  // only divergent point: guarded store on the tail tile
            float o = sTv[m * 16 + r * 4 + 3];
#pragma unroll
            for (int c = 0; c < 3; ++c)
                o += sTv[m * 16 + r * 4 + c] * svp[m * 3 + c];
            out[v * 3 + r] = o + Th[r];
        }
    }
}

// ---------------------------------------------------------------------------
extern "C" void kernel_launch(void* const* d_in, const int* in_sizes, int n_in,
                              void* d_out, int out_size, void* d_ws, size_t ws_size,
                              hipStream_t stream) {
    const float* poses  = (const float*)d_in[0];  // (512,72)  -> use batch 0
    const float* shapes = (const float*)d_in[1];  // (512,10)  -> use batch 0
    const float* Rh     = (const float*)d_in[2];  // (512,3)   -> use batch 0
    const float* Th     = (const float*)d_in[3];  // (512,3)   -> use batch 0
    const float* vt     = (const float*)d_in[4];  // (6890,3)
    const float* sd     = (const float*)d_in[5];  // (6890,3,10)
    const float* pdirs  = (const float*)d_in[6];  // (207, 20670)
    const float* JR     = (const float*)d_in[7];  // (24, 6890)
    const float* W      = (const float*)d_in[8];  // (6890, 24)
    float* out = (float*)d_out;

    float* ws      = (float*)d_ws;
    float* vshaped = ws;          // 20670 floats
    float* Jws     = ws + 20672;  // 72 floats
    float* pfw     = ws + 20768;  // 207 floats
    float* Aws     = ws + 20992;  // 24*16 floats

    k_vshaped<<<(NV3 + 255) / 256, 256, 0, stream>>>(vt, sd, shapes, vshaped);
    k_joints<<<24, 256, 0, stream>>>(JR, vshaped, Jws);
    k_chain<<<1, 32, 0, stream>>>(poses, Rh, Jws, pfw, Aws);
    int nTiles = (NV + 15) / 16;  // 431
    k_lbs<<<(nTiles + 3) / 4, 128, 0, stream>>>(W, Aws, pfw, pdirs, vshaped, Th, out, nTiles);
}